// BipartiteAttentionLayer_75290776698914
// MI455X (gfx1250) — compile-verified
//
#include <hip/hip_runtime.h>
#include <hip/hip_bf16.h>

typedef __attribute__((ext_vector_type(2))) float v2f;
typedef __attribute__((ext_vector_type(8))) float v8f;

#define INPUT_DIM 256
#define HIDDEN_DIM 128
#define N_S 2048
#define N_C 1024

static __device__ __forceinline__ float fast_tanh(float x) {
#if __has_builtin(__builtin_amdgcn_tanhf)
  return __builtin_amdgcn_tanhf(x);   // V_TANH_F32 (TRANS) on gfx1250
#else
  return tanhf(x);
#endif
}

// -------------------------------------------------------------------------
// Generic f32 WMMA GEMM: D[M,N] = A[M,K] @ B[K,N] (+ bias[N] if use_bias)
// One wave32 per 16x16 output tile, K-loop in steps of 4 using
// V_WMMA_F32_16X16X4_F32.
//
// A 16x4 f32 lane layout (ISA 7.12.2): lane L holds row M=L%16,
//   VGPR0/1 = K = 2*(L/16) and 2*(L/16)+1  -> one float2 load (contiguous).
// B 4x16 mirrored: VGPR r holds row K = 2*(L/16)+r, col N = L%16.
// C/D 16x16 f32: VGPR r, lane L -> M = r + 8*(L/16), N = L%16.
// -------------------------------------------------------------------------
__global__ void wmma_gemm_bias_f32(const float* __restrict__ A,
                                   const float* __restrict__ B,
                                   const float* __restrict__ bias,
                                   float* __restrict__ D,
                                   int M, int N, int K, int use_bias) {
  const int ntn  = N >> 4;
  const int tile = blockIdx.x;
  const int tm   = tile / ntn;
  const int tn   = tile - tm * ntn;

  const int lane = threadIdx.x & 31;
  const int mrow = lane & 15;       // M within tile
  const int half = lane >> 4;       // selects K pair / upper M half
  const int ncol = (tn << 4) + (lane & 15);

  const float* __restrict__ arow = A + (size_t)((tm << 4) + mrow) * K + 2 * half;
  const float* __restrict__ bcol = B + (size_t)(2 * half) * N + ncol;

  v8f acc = {};
  for (int k = 0; k < K; k += 4) {
    v2f a;
    a.x = arow[k];
    a.y = arow[k + 1];
    v2f b;
    b.x = bcol[(size_t)k * N];
    b.y = bcol[(size_t)(k + 1) * N];
    acc = __builtin_amdgcn_wmma_f32_16x16x4_f32(
        /*neg_a=*/false, a, /*neg_b=*/false, b,
        /*c_mod=*/(short)0, acc, /*reuse_a=*/false, /*reuse_b=*/false);
  }

  float bv = 0.0f;
  if (use_bias) bv = bias[ncol];

  float* __restrict__ drow = D + (size_t)((tm << 4) + 8 * half) * N + ncol;
#pragma unroll
  for (int r = 0; r < 8; ++r) {
    drow[(size_t)r * N] = acc[r] + bv;
  }
}

// -------------------------------------------------------------------------
// Fused additive-attention score + softmax.
// One 256-thread block per student row s:
//   score[c] = sum_h w_a[h] * tanh(S[s,h] + C[c,h]) + b_a
//   weights[s,:] = softmax(score)
// S-row and w_a staged in LDS; C (512 KB total) is L2-resident.
// -------------------------------------------------------------------------
__global__ void score_softmax_kernel(const float* __restrict__ Sbuf,
                                     const float* __restrict__ Cbuf,
                                     const float* __restrict__ w_a,
                                     const float* __restrict__ b_a,
                                     float* __restrict__ Wout) {
  __shared__ float s_row[HIDDEN_DIM];
  __shared__ float s_wa[HIDDEN_DIM];
  __shared__ float sc[N_C];
  __shared__ float red[256];

  const int s   = blockIdx.x;
  const int tid = threadIdx.x;

  if (tid < HIDDEN_DIM) {
    s_row[tid] = Sbuf[(size_t)s * HIDDEN_DIM + tid];
    s_wa[tid]  = w_a[tid];
  }
  __syncthreads();

  const float badd = b_a[0];

  // ---- scores: each thread owns one college per 256-wide chunk ----
  for (int base = 0; base < N_C; base += 256) {
    const int c = base + tid;
    const float4* __restrict__ crow =
        (const float4*)(Cbuf + (size_t)c * HIDDEN_DIM);
    float acc = 0.0f;
#pragma unroll 8
    for (int h4 = 0; h4 < HIDDEN_DIM / 4; ++h4) {
      const float4 cv = crow[h4];
      const float4 sv = *(const float4*)(s_row + 4 * h4);
      const float4 wv = *(const float4*)(s_wa + 4 * h4);
      acc += wv.x * fast_tanh(sv.x + cv.x);
      acc += wv.y * fast_tanh(sv.y + cv.y);
      acc += wv.z * fast_tanh(sv.z + cv.z);
      acc += wv.w * fast_tanh(sv.w + cv.w);
    }
    sc[c] = acc + badd;
  }
  __syncthreads();

  // ---- row max ----
  float m = -3.0e38f;
  for (int c = tid; c < N_C; c += 256) m = fmaxf(m, sc[c]);
  red[tid] = m;
  __syncthreads();
  for (int off = 128; off > 0; off >>= 1) {
    if (tid < off) red[tid] = fmaxf(red[tid], red[tid + off]);
    __syncthreads();
  }
  m = red[0];
  __syncthreads();

  // ---- exp + row sum ----
  float sum = 0.0f;
  for (int c = tid; c < N_C; c += 256) {
    const float e = __expf(sc[c] - m);
    sc[c] = e;
    sum += e;
  }
  __syncthreads();
  red[tid] = sum;
  __syncthreads();
  for (int off = 128; off > 0; off >>= 1) {
    if (tid < off) red[tid] += red[tid + off];
    __syncthreads();
  }
  const float inv = 1.0f / red[0];

  // ---- normalized weights -> global ----
  for (int c = tid; c < N_C; c += 256) {
    Wout[(size_t)s * N_C + c] = sc[c] * inv;
  }
}

extern "C" void kernel_launch(void* const* d_in, const int* in_sizes, int n_in,
                              void* d_out, int out_size, void* d_ws, size_t ws_size,
                              hipStream_t stream) {
  const float* student = (const float*)d_in[0];  // [N_S, INPUT_DIM]
  const float* college = (const float*)d_in[1];  // [N_C, INPUT_DIM]
  const float* W_s     = (const float*)d_in[2];  // [INPUT_DIM, HIDDEN_DIM]
  const float* b_s     = (const float*)d_in[3];  // [HIDDEN_DIM]
  const float* W_c     = (const float*)d_in[4];
  const float* b_c     = (const float*)d_in[5];
  const float* w_a     = (const float*)d_in[6];  // [HIDDEN_DIM]
  const float* b_a     = (const float*)d_in[7];  // scalar

  float* attended = (float*)d_out;                          // [N_S, INPUT_DIM]
  float* weights  = (float*)d_out + (size_t)N_S * INPUT_DIM; // [N_S, N_C]

  float* Sbuf = (float*)d_ws;                       // [N_S, HIDDEN_DIM]
  float* Cbuf = Sbuf + (size_t)N_S * HIDDEN_DIM;    // [N_C, HIDDEN_DIM]

  // 1) S = student @ W_s + b_s   (M=2048, N=128, K=256)
  wmma_gemm_bias_f32<<<(N_S / 16) * (HIDDEN_DIM / 16), 32, 0, stream>>>(
      student, W_s, b_s, Sbuf, N_S, HIDDEN_DIM, INPUT_DIM, 1);

  // 2) C = college @ W_c + b_c   (M=1024, N=128, K=256)
  wmma_gemm_bias_f32<<<(N_C / 16) * (HIDDEN_DIM / 16), 32, 0, stream>>>(
      college, W_c, b_c, Cbuf, N_C, HIDDEN_DIM, INPUT_DIM, 1);

  // 3) scores + softmax -> weights (second output region)
  score_softmax_kernel<<<N_S, 256, 0, stream>>>(Sbuf, Cbuf, w_a, b_a, weights);

  // 4) attended = weights @ college_feats  (M=2048, N=256, K=1024)
  wmma_gemm_bias_f32<<<(N_S / 16) * (INPUT_DIM / 16), 32, 0, stream>>>(
      weights, college, nullptr, attended, N_S, INPUT_DIM, N_C, 0);
}